// QwenPixelBridge_4312147165307
// MI455X (gfx1250) — compile-verified
//
#include <hip/hip_runtime.h>
#include <hip/hip_bf16.h>
#include <math.h>

typedef float v2f __attribute__((ext_vector_type(2)));
typedef float v8f __attribute__((ext_vector_type(8)));

// Problem constants (match setup_inputs: B=8,K=32,H=W=256,C=512,Dt=2048,m=5,ring=7)
constexpr int Bn = 8, Kn = 32, Hn = 256, Wn = 256, Cn = 512, Dt = 2048;
constexpr int Mm = 5;              // max_masks
constexpr int HWn = Hn * Wn;       // 65536
constexpr int ROWS = Bn * 2 * Mm;  // 80 pooled rows

__device__ __forceinline__ float sigmoidf_(float x) {
    return 1.0f / (1.0f + __expf(-x));
}

// ---------------- Z: zero accumulators (ws is poisoned) ----------------
__global__ void zero_kernel(float* __restrict__ praw, float* __restrict__ rsum) {
    int i = blockIdx.x * 256 + threadIdx.x;
    if (i < ROWS * Cn) praw[i] = 0.0f;
    if (i < Bn * Mm)   rsum[i] = 0.0f;
}

// ---------------- A: areas[b,k] = sum_hw sigmoid(logits) ----------------
__global__ void areas_kernel(const float* __restrict__ logits, float* __restrict__ areas) {
    int bk = blockIdx.x;                         // 0..255
    const float* p = logits + (size_t)bk * HWn;
    float s = 0.0f;
    for (int i = threadIdx.x; i < HWn; i += 256) s += sigmoidf_(p[i]);
    __shared__ float red[256];
    red[threadIdx.x] = s;
    __syncthreads();
    for (int st = 128; st > 0; st >>= 1) {
        if (threadIdx.x < st) red[threadIdx.x] += red[threadIdx.x + st];
        __syncthreads();
    }
    if (threadIdx.x == 0) areas[bk] = red[0];
}

// ---------------- B: stable top-5 per batch + core norms ----------------
__global__ void topk_kernel(const float* __restrict__ areas,
                            int* __restrict__ selIdx, float* __restrict__ norms) {
    int b = blockIdx.x;
    if (threadIdx.x != 0) return;
    float a[Kn];
    bool used[Kn];
    for (int k = 0; k < Kn; ++k) { a[k] = areas[b * Kn + k]; used[k] = false; }
    for (int j = 0; j < Mm; ++j) {
        int best = 0; float bv = -INFINITY;
        for (int k = 0; k < Kn; ++k)
            if (!used[k] && a[k] > bv) { bv = a[k]; best = k; }   // strict > => lowest index wins ties
        used[best] = true;
        selIdx[b * Mm + j] = best;
        norms[b * 2 * Mm + j] = fmaxf(bv, 1e-6f);                 // core norm == clipped area
    }
}

// ---------------- C: selected sigmoid + 7x7 dilation ring ----------------
__global__ void ring_kernel_fn(const float* __restrict__ logits, const int* __restrict__ selIdx,
                               float* __restrict__ S, float* __restrict__ R,
                               float* __restrict__ ringsum) {
    int z = blockIdx.z;                 // 0..39 -> (b,j)
    int b = z / Mm, /*j = z % Mm,*/ sel = selIdx[z];
    const float* src = logits + (size_t)(b * Kn + sel) * HWn;
    __shared__ float t[22][24];
    __shared__ float acc;
    int tid = threadIdx.y * 16 + threadIdx.x;
    if (tid == 0) acc = 0.0f;
    int ox = blockIdx.x * 16, oy = blockIdx.y * 16;
    for (int e = tid; e < 22 * 22; e += 256) {
        int r = e / 22, c = e % 22;
        int gy = oy + r - 3, gx = ox + c - 3;
        float v = -INFINITY;                                     // 'SAME' maxpool pad
        if (gy >= 0 && gy < Hn && gx >= 0 && gx < Wn)
            v = sigmoidf_(src[gy * Wn + gx]);
        t[r][c] = v;
    }
    __syncthreads();
    int tx = threadIdx.x, ty = threadIdx.y;
    float s = t[ty + 3][tx + 3];
    float d = -INFINITY;
#pragma unroll
    for (int dy = 0; dy < 7; ++dy)
#pragma unroll
        for (int dx = 0; dx < 7; ++dx)
            d = fmaxf(d, t[ty + dy][tx + dx]);
    float ring = fmaxf(d - s, 0.0f);
    size_t o = (size_t)z * HWn + (size_t)(oy + ty) * Wn + (ox + tx);
    S[o] = s;
    R[o] = ring;
    atomicAdd(&acc, ring);              // ds_add_f32
    __syncthreads();
    if (tid == 0) atomicAdd(&ringsum[z], acc);
}

// ---------------- D: context norms + empty-ring additive constant ----------------
__global__ void ctxnorm_kernel(const float* __restrict__ ringsum,
                               float* __restrict__ norms, float* __restrict__ emptyAdd) {
    int i = threadIdx.x;
    if (i >= Bn * Mm) return;
    int b = i / Mm, j = i % Mm;
    float rs = ringsum[i];
    float add = (rs == 0.0f) ? 1e-4f : 0.0f;     // ring += empty * 1e-4 (uniform)
    emptyAdd[i] = add;
    norms[b * 2 * Mm + Mm + j] = fmaxf(rs + add * (float)HWn, 1e-6f);
}

// ---------------- E: weighted pooling via V_WMMA_F32_16X16X4_F32 ----------------
// D[16c x 16j] += A(feat 16c x 4hw) * B(weights 4hw x 16j), accumulated over hw.
// grid = (32 hw-chunks, 4 c-groups, 8 batches), block = 256 (8 waves, 16 c each).
__global__ __launch_bounds__(256) void wpool_kernel(
        const float* __restrict__ feat, const float* __restrict__ S,
        const float* __restrict__ R,    const float* __restrict__ emptyAdd,
        float* __restrict__ pooledRaw) {
    int b = blockIdx.z, cg = blockIdx.y, chunk = blockIdx.x;
    int tid = threadIdx.x, wave = tid >> 5, lane = tid & 31;
    int cBase = cg * 128 + wave * 16;
    int hwBase = chunk * 2048;

    __shared__ float wt[16][68];          // [j][hw_local], pad 68 -> conflict-free b64 reads

    v8f acc = {};
    const float* fbase = feat + (size_t)b * Cn * HWn;
    int m   = lane & 15;                  // c row within tile / j col for B
    int kh  = (lane >> 4) * 2;            // K sub-rows handled by this half-wave
    const float* frow = fbase + (size_t)(cBase + m) * HWn;

    for (int s0 = 0; s0 < 2048; s0 += 64) {
        int hw0 = hwBase + s0;
        __syncthreads();
        // stage weights (rows 0-4: core S, 5-9: ring R + emptyAdd, 10-15: zero)
        for (int e = tid; e < 1024; e += 256) {
            int j = e >> 6, hl = e & 63;
            float w = 0.0f;
            if (j < Mm)
                w = S[(size_t)(b * Mm + j) * HWn + hw0 + hl];
            else if (j < 2 * Mm)
                w = R[(size_t)(b * Mm + (j - Mm)) * HWn + hw0 + hl] + emptyAdd[b * Mm + (j - Mm)];
            wt[j][hl] = w;
        }
        __syncthreads();

        const float* fp = frow + hw0 + kh;
#pragma unroll
        for (int kk = 0; kk < 16; ++kk) {
            // A fragment: lane m holds row c, halves cover K pairs (ISA 16x4 f32 layout)
            v2f a = *(const v2f*)(fp + kk * 4);
            // B fragment: col j = lane&15, rows K = kh, kh+1
            v2f bb = *(const v2f*)&wt[m][kk * 4 + kh];
            acc = __builtin_amdgcn_wmma_f32_16x16x4_f32(
                false, a, false, bb, (short)0, acc, false, false);
        }
    }

    // D layout: VGPR v -> M = v (lanes 0-15) / v+8 (lanes 16-31); N = lane&15 = j
    if (m < 2 * Mm) {
        float* dst = pooledRaw + (size_t)(b * 2 * Mm + m) * Cn + cBase;
#pragma unroll
        for (int v = 0; v < 8; ++v) {
            int cl = (lane < 16) ? v : v + 8;
            atomicAdd(&dst[cl], acc[v]);
        }
    }
}

// ---------------- F0: normalize pooled rows ----------------
__global__ void norm_kernel(const float* __restrict__ praw, const float* __restrict__ norms,
                            float* __restrict__ pooled) {
    int i = blockIdx.x * 256 + threadIdx.x;
    if (i >= ROWS * Cn) return;
    pooled[i] = praw[i] / norms[i >> 9];       // Cn = 512
}

// ---------------- F1: projection [80,512] @ [512,2048]^T + bias, via WMMA ----------------
// grid = (Dt/16 = 128, ROWS/16 = 5), block = 32 (one wave).
__global__ void proj_kernel(const float* __restrict__ pooled, const float* __restrict__ Wm,
                            const float* __restrict__ bias, float* __restrict__ out) {
    int lane = threadIdx.x & 31;
    int rBase = blockIdx.y * 16;
    int dBase = blockIdx.x * 16;
    int m  = lane & 15;
    int kh = (lane >> 4) * 2;
    const float* arow = pooled + (size_t)(rBase + m) * Cn + kh;   // A: row r, K pairs
    const float* brow = Wm + (size_t)(dBase + m) * Cn + kh;       // B: col d, K pairs (proj_w[d][c])
    v8f acc = {};
#pragma unroll 4
    for (int k = 0; k < Cn; k += 4) {
        v2f a  = *(const v2f*)(arow + k);
        v2f bb = *(const v2f*)(brow + k);
        acc = __builtin_amdgcn_wmma_f32_16x16x4_f32(
            false, a, false, bb, (short)0, acc, false, false);
    }
    float bv = bias[dBase + m];
#pragma unroll
    for (int v = 0; v < 8; ++v) {
        int r = rBase + ((lane < 16) ? v : v + 8);
        out[(size_t)r * Dt + dBase + m] = acc[v] + bv;
    }
}

// ---------------- host launcher ----------------
extern "C" void kernel_launch(void* const* d_in, const int* in_sizes, int n_in,
                              void* d_out, int out_size, void* d_ws, size_t ws_size,
                              hipStream_t stream) {
    (void)in_sizes; (void)n_in; (void)out_size; (void)ws_size;
    const float* mask_logits = (const float*)d_in[0];   // [8,32,256,256]
    const float* pixel_feat  = (const float*)d_in[1];   // [8,512,256,256]
    const float* proj_w      = (const float*)d_in[2];   // [2048,512]
    const float* proj_b      = (const float*)d_in[3];   // [2048]
    // d_in[4]=max_masks(5), d_in[5]=ring_kernel(7): compiled in as constants.
    float* out = (float*)d_out;                         // [8,10,2048]

    char* ws = (char*)d_ws;
    const size_t nSR = (size_t)Bn * Mm * HWn * sizeof(float);     // 10,485,760 B each
    float* Sbuf   = (float*)(ws);
    float* Rbuf   = (float*)(ws + nSR);
    float* areas  = (float*)(ws + 2 * nSR);                        // 256 f
    int*   selIdx = (int*)  (ws + 2 * nSR + 1024);                 // 40 i
    float* norms  = (float*)(ws + 2 * nSR + 1280);                 // 80 f
    float* eAdd   = (float*)(ws + 2 * nSR + 1792);                 // 40 f
    float* rsum   = (float*)(ws + 2 * nSR + 2048);                 // 40 f
    float* praw   = (float*)(ws + 2 * nSR + 2304);                 // 40960 f
    float* pooled = (float*)(ws + 2 * nSR + 2304 + (size_t)ROWS * Cn * sizeof(float));

    zero_kernel<<<(ROWS * Cn + 255) / 256, 256, 0, stream>>>(praw, rsum);
    areas_kernel<<<Bn * Kn, 256, 0, stream>>>(mask_logits, areas);
    topk_kernel<<<Bn, 32, 0, stream>>>(areas, selIdx, norms);
    ring_kernel_fn<<<dim3(Wn / 16, Hn / 16, Bn * Mm), dim3(16, 16), 0, stream>>>(
        mask_logits, selIdx, Sbuf, Rbuf, rsum);
    ctxnorm_kernel<<<1, 64, 0, stream>>>(rsum, norms, eAdd);
    wpool_kernel<<<dim3(32, 4, Bn), 256, 0, stream>>>(pixel_feat, Sbuf, Rbuf, eAdd, praw);
    norm_kernel<<<(ROWS * Cn + 255) / 256, 256, 0, stream>>>(praw, norms, pooled);
    proj_kernel<<<dim3(Dt / 16, ROWS / 16), 32, 0, stream>>>(pooled, proj_w, proj_b, out);
}